// Subtask1Model_9483287790255
// MI455X (gfx1250) — compile-verified
//
#include <hip/hip_runtime.h>
#include <hip/hip_bf16.h>

typedef __attribute__((ext_vector_type(2))) float v2f;
typedef __attribute__((ext_vector_type(8))) float v8f;

#define ALPHA 0.1f

static __device__ __forceinline__ v8f wmma4(v2f a, v2f b, v8f c) {
    return __builtin_amdgcn_wmma_f32_16x16x4_f32(false, a, false, b,
                                                 (short)0, c, false, false);
}

// ---------------------------------------------------------------------------
// Kernel 1: ut_embed[bl, 0..799] = concat(mean_f(word_table[words[b,l,0,f]]),
//                                         spk_table[speakers[b,l]])
// ---------------------------------------------------------------------------
__global__ void embed_kernel(const int* __restrict__ words,
                             const int* __restrict__ speakers,
                             const float* __restrict__ word_table,
                             const float* __restrict__ spk_table,
                             float* __restrict__ ut_embed) {
    int bl = blockIdx.x;                 // 0..511  (b*32 + l)
    int i0 = bl * 128;                   // ((b*32+l)*32 + u=0)*4
    int w0 = words[i0 + 0];
    int w1 = words[i0 + 1];
    int w2 = words[i0 + 2];
    int w3 = words[i0 + 3];
    int spk = speakers[bl];
    for (int c = threadIdx.x; c < 800; c += blockDim.x) {
        float v;
        if (c < 768) {
            v = 0.25f * (word_table[(size_t)w0 * 768 + c] +
                         word_table[(size_t)w1 * 768 + c] +
                         word_table[(size_t)w2 * 768 + c] +
                         word_table[(size_t)w3 * 768 + c]);
        } else {
            v = spk_table[spk * 32 + (c - 768)];
        }
        ut_embed[(size_t)bl * 800 + c] = v;
    }
}

// ---------------------------------------------------------------------------
// Generic fp32 WMMA GEMM: C[M,N] = act(A[M,K] @ B[K,N] + bias)
// one wave computes a 16x64 strip (4 accumulators), sharing the A fragment
// across 4 WMMAs per K-step.  V_WMMA_F32_16X16X4_F32.
// grid = (N/64, M/16), block = 32.
// ---------------------------------------------------------------------------
__global__ __launch_bounds__(32) void gemm_f32_wmma(
    const float* __restrict__ A, const float* __restrict__ B,
    const float* __restrict__ bias, float* __restrict__ C,
    int K, int lda, int ldb, int ldc, int act) {
    int nt0 = blockIdx.x * 4;        // first of 4 consecutive N tiles
    int mt  = blockIdx.y;            // M tile
    int lane = threadIdx.x & 31;
    int half = lane >> 4;            // 0: K=0,1   1: K=2,3
    int l16  = lane & 15;

    const float* Arow = A + (size_t)(mt * 16 + l16) * lda;
    const float* Bcol = B + nt0 * 16 + l16;

    v8f c0 = {}, c1 = {}, c2 = {}, c3 = {};
    #pragma unroll 2
    for (int k = 0; k < K; k += 4) {
        int kk = k + 2 * half;
        v2f a;
        a.x = Arow[kk];
        a.y = Arow[kk + 1];
        const float* bp0 = Bcol + (size_t)kk * ldb;
        const float* bp1 = Bcol + (size_t)(kk + 1) * ldb;
        v2f b0, b1, b2, b3;
        b0.x = bp0[0];  b0.y = bp1[0];
        b1.x = bp0[16]; b1.y = bp1[16];
        b2.x = bp0[32]; b2.y = bp1[32];
        b3.x = bp0[48]; b3.y = bp1[48];
        c0 = wmma4(a, b0, c0);
        c1 = wmma4(a, b1, c1);
        c2 = wmma4(a, b2, c2);
        c3 = wmma4(a, b3, c3);
    }

    v8f acc[4] = {c0, c1, c2, c3};
    #pragma unroll
    for (int t = 0; t < 4; ++t) {
        #pragma unroll
        for (int r = 0; r < 8; ++r) {
            int m = mt * 16 + r + 8 * half;
            int n = (nt0 + t) * 16 + l16;
            float v = acc[t][r];
            if (bias) v += bias[n];
            if (act) v = (v > 0.f) ? v : ALPHA * v;
            C[(size_t)m * ldc + n] = v;
        }
    }
}

// ---------------------------------------------------------------------------
// Rank-1 biaffine terms: xu[b,o,i] = X[b,i,:]·W[o,:,512]
//                        vy[b,o,i] = Y[b,i,:]·W[o,512,:]
// ---------------------------------------------------------------------------
__global__ void rank1_kernel(const float* __restrict__ X,
                             const float* __restrict__ Y,
                             const float* __restrict__ Wattn,
                             float* __restrict__ xu, float* __restrict__ vy,
                             int O, int total) {
    int t = blockIdx.x * blockDim.x + threadIdx.x;
    if (t >= total) return;
    int i  = t & 31;
    int bo = t >> 5;
    int o  = bo % O;
    int b  = bo / O;
    const float* W  = Wattn + (size_t)o * 513 * 513;
    const float* xr = X + ((size_t)b * 32 + i) * 512;
    const float* yr = Y + ((size_t)b * 32 + i) * 512;
    float su = 0.f, sv = 0.f;
    for (int k = 0; k < 512; ++k) {
        su += xr[k] * W[(size_t)k * 513 + 512];
        sv += yr[k] * W[(size_t)512 * 513 + k];
    }
    xu[t] = su;
    vy[t] = sv;
}

// ---------------------------------------------------------------------------
// Fused biaffine per (b,o): S = X_b A_o Y_b^T + xu + vy + c, scattered into
// out[((b*32+i)*32+j)*O + o].
// Phase A: P = A_o @ Y_b^T (512x32) into 64 KB LDS; each wave computes a
//          64x16 strip (4 M-tiles sharing one Y fragment per K-step).
// Phase B: S = X_b @ P (32x32) on waves 0..3.
// ---------------------------------------------------------------------------
__global__ __launch_bounds__(256) void biaffine_wmma(
    const float* __restrict__ X, const float* __restrict__ Y,
    const float* __restrict__ Wattn,
    const float* __restrict__ xu, const float* __restrict__ vy,
    float* __restrict__ out, int O) {
    __shared__ float P[512 * 32];
    int bo = blockIdx.x;
    int b = bo / O, o = bo % O;
    int tid  = threadIdx.x;
    int wave = tid >> 5;
    int lane = tid & 31;
    int half = lane >> 4;
    int l16  = lane & 15;

    const float* W  = Wattn + (size_t)o * 513 * 513;   // A = W[0:512,0:512], stride 513
    const float* Xb = X + (size_t)b * 32 * 512;
    const float* Yb = Y + (size_t)b * 32 * 512;

    // Phase A: P[r][n] = sum_j A[r][j] * Y[n][j]
    // wave w owns M rows [w*64, w*64+64); loop over the two 16-col groups of P
    {
        int m0 = wave * 64;
        const float* Ar0 = W + (size_t)(m0 +  0 + l16) * 513;
        const float* Ar1 = W + (size_t)(m0 + 16 + l16) * 513;
        const float* Ar2 = W + (size_t)(m0 + 32 + l16) * 513;
        const float* Ar3 = W + (size_t)(m0 + 48 + l16) * 513;
        #pragma unroll
        for (int ntl = 0; ntl < 2; ++ntl) {
            const float* Yr = Yb + (size_t)(ntl * 16 + l16) * 512;
            v8f c0 = {}, c1 = {}, c2 = {}, c3 = {};
            #pragma unroll 2
            for (int k = 0; k < 512; k += 4) {
                int kk = k + 2 * half;
                v2f bb;
                bb.x = Yr[kk];
                bb.y = Yr[kk + 1];
                v2f a0, a1, a2, a3;
                a0.x = Ar0[kk]; a0.y = Ar0[kk + 1];
                a1.x = Ar1[kk]; a1.y = Ar1[kk + 1];
                a2.x = Ar2[kk]; a2.y = Ar2[kk + 1];
                a3.x = Ar3[kk]; a3.y = Ar3[kk + 1];
                c0 = wmma4(a0, bb, c0);
                c1 = wmma4(a1, bb, c1);
                c2 = wmma4(a2, bb, c2);
                c3 = wmma4(a3, bb, c3);
            }
            v8f acc[4] = {c0, c1, c2, c3};
            #pragma unroll
            for (int t = 0; t < 4; ++t) {
                #pragma unroll
                for (int r = 0; r < 8; ++r) {
                    int m = m0 + t * 16 + r + 8 * half;
                    int n = ntl * 16 + l16;
                    P[m * 32 + n] = acc[t][r];
                }
            }
        }
    }
    __syncthreads();

    // Phase B: S[i][j] = sum_r X[i][r] * P[r][j]; 4 tiles on waves 0..3
    if (wave < 4) {
        int mt  = wave >> 1;
        int ntl = wave & 1;
        const float* Xr = Xb + (size_t)(mt * 16 + l16) * 512;
        int pcol = ntl * 16 + l16;
        v8f c = {};
        #pragma unroll 4
        for (int k = 0; k < 512; k += 4) {
            int kk = k + 2 * half;
            v2f a, bb;
            a.x  = Xr[kk];
            a.y  = Xr[kk + 1];
            bb.x = P[kk * 32 + pcol];
            bb.y = P[(kk + 1) * 32 + pcol];
            c = wmma4(a, bb, c);
        }
        float cc = W[(size_t)512 * 513 + 512];
        #pragma unroll
        for (int r = 0; r < 8; ++r) {
            int i = mt * 16 + r + 8 * half;
            int j = ntl * 16 + l16;
            float v = c[r] + xu[bo * 32 + i] + vy[bo * 32 + j] + cc;
            out[(((size_t)b * 32 + i) * 32 + j) * O + o] = v;
        }
    }
}

// ---------------------------------------------------------------------------
// sc[r] = leaky(h2[r,:]·span_W + span_b)
// ---------------------------------------------------------------------------
__global__ void span_score_kernel(const float* __restrict__ h2,
                                  const float* __restrict__ span_W,
                                  const float* __restrict__ span_b,
                                  float* __restrict__ sc) {
    int r = blockIdx.x * blockDim.x + threadIdx.x;
    if (r >= 512) return;
    float s = span_b[0];
    for (int k = 0; k < 768; ++k) s += h2[(size_t)r * 768 + k] * span_W[k];
    sc[r] = (s > 0.f) ? s : ALPHA * s;
}

// ---------------------------------------------------------------------------
// s_span[b,i,j,u] = mask[b,i,j] ? sc[b,i] : -1.0
// ---------------------------------------------------------------------------
__global__ void span_fill_kernel(const int* __restrict__ graphs,
                                 const unsigned char* __restrict__ pad,
                                 const float* __restrict__ sc,
                                 float* __restrict__ out) {
    int t = blockIdx.x * blockDim.x + threadIdx.x;
    if (t >= 16 * 32 * 32 * 31) return;
    int bij = t / 31;                  // ((b*32+i)*32+j)
    int j  = bij & 31;
    int bi = bij >> 5;
    int i  = bi & 31;
    int b  = bi >> 5;
    bool m = (graphs[bij] != 0) && (pad[b * 32 + i] != 0) && (pad[b * 32 + j] != 0);
    out[t] = m ? sc[b * 32 + i] : -1.0f;
}

// ---------------------------------------------------------------------------
extern "C" void kernel_launch(void* const* d_in, const int* in_sizes, int n_in,
                              void* d_out, int out_size, void* d_ws, size_t ws_size,
                              hipStream_t stream) {
    const int*   words      = (const int*)d_in[0];
    const int*   speakers   = (const int*)d_in[1];
    const unsigned char* pad = (const unsigned char*)d_in[2];
    const int*   graphs     = (const int*)d_in[3];
    const float* word_table = (const float*)d_in[4];
    const float* spk_table  = (const float*)d_in[5];
    const float* ut_cause_W  = (const float*)d_in[6];
    const float* ut_cause_b  = (const float*)d_in[7];
    const float* ut_effect_W = (const float*)d_in[8];
    const float* ut_effect_b = (const float*)d_in[9];
    const float* em_cause_W  = (const float*)d_in[10];
    const float* em_cause_b  = (const float*)d_in[11];
    const float* em_effect_W = (const float*)d_in[12];
    const float* em_effect_b = (const float*)d_in[13];
    const float* ut_attn_W   = (const float*)d_in[14];
    const float* em_attn_W   = (const float*)d_in[15];
    const float* v_proj_W    = (const float*)d_in[16];
    const float* v_proj_b    = (const float*)d_in[17];
    const float* out_proj_W  = (const float*)d_in[18];
    const float* out_proj_b  = (const float*)d_in[19];
    const float* span_W      = (const float*)d_in[20];
    const float* span_b      = (const float*)d_in[21];

    // workspace layout (fp32)
    float* ws       = (float*)d_ws;
    float* ut_embed = ws;                    // 512*800   = 409600
    float* uc       = ut_embed + 409600;     // 512*512
    float* ue       = uc + 262144;
    float* ec       = ue + 262144;
    float* ee       = ec + 262144;
    float* xu_ut    = ee + 262144;           // 16*2*32 = 1024
    float* vy_ut    = xu_ut + 1024;
    float* xu_em    = vy_ut + 1024;          // 16*7*32 = 3584
    float* vy_em    = xu_em + 3584;
    float* h1       = vy_em + 3584;          // 512*768
    float* h2       = h1 + 393216;           // 512*768
    float* sc       = h2 + 393216;           // 512

    float* out    = (float*)d_out;
    float* s_ut   = out;                     // (16,32,32,2)  = 32768
    float* s_em   = out + 32768;             // (16,32,32,7)  = 114688
    float* s_span = out + 147456;            // (16,32,32,31) = 507904

    // 1. embeddings
    embed_kernel<<<512, 256, 0, stream>>>(words, speakers, word_table, spk_table,
                                          ut_embed);

    // 2. four FFNs: (512x800)@(800x512) + b, leaky
    dim3 gf(512 / 64, 512 / 16);
    gemm_f32_wmma<<<gf, 32, 0, stream>>>(ut_embed, ut_cause_W,  ut_cause_b,  uc,
                                         800, 800, 512, 512, 1);
    gemm_f32_wmma<<<gf, 32, 0, stream>>>(ut_embed, ut_effect_W, ut_effect_b, ue,
                                         800, 800, 512, 512, 1);
    gemm_f32_wmma<<<gf, 32, 0, stream>>>(ut_embed, em_cause_W,  em_cause_b,  ec,
                                         800, 800, 512, 512, 1);
    gemm_f32_wmma<<<gf, 32, 0, stream>>>(ut_embed, em_effect_W, em_effect_b, ee,
                                         800, 800, 512, 512, 1);

    // 3. rank-1 biaffine terms
    rank1_kernel<<<(16 * 2 * 32 + 255) / 256, 256, 0, stream>>>(
        uc, ue, ut_attn_W, xu_ut, vy_ut, 2, 16 * 2 * 32);
    rank1_kernel<<<(16 * 7 * 32 + 255) / 256, 256, 0, stream>>>(
        ec, ee, em_attn_W, xu_em, vy_em, 7, 16 * 7 * 32);

    // 4. fused biaffines -> s_ut, s_em (written directly to d_out)
    biaffine_wmma<<<16 * 2, 256, 0, stream>>>(uc, ue, ut_attn_W, xu_ut, vy_ut,
                                              s_ut, 2);
    biaffine_wmma<<<16 * 7, 256, 0, stream>>>(ec, ee, em_attn_W, xu_em, vy_em,
                                              s_em, 7);

    // 5. attn_out = (ee @ v_proj + b) @ out_proj + b
    gemm_f32_wmma<<<dim3(768 / 64, 512 / 16), 32, 0, stream>>>(
        ee, v_proj_W, v_proj_b, h1, 512, 512, 768, 768, 0);
    gemm_f32_wmma<<<dim3(768 / 64, 512 / 16), 32, 0, stream>>>(
        h1, out_proj_W, out_proj_b, h2, 768, 768, 768, 768, 0);

    // 6. span score + masked broadcast
    span_score_kernel<<<2, 256, 0, stream>>>(h2, span_W, span_b, sc);
    span_fill_kernel<<<(16 * 32 * 32 * 31 + 255) / 256, 256, 0, stream>>>(
        graphs, pad, sc, s_span);
}